// WavetableSynth_55301998903307
// MI455X (gfx1250) — compile-verified
//
#include <hip/hip_runtime.h>
#include <stdint.h>

#define SRATE      16000
#define T_SAMPLES  960000
#define N_TAB      64
#define WT_LEN     512
#define FRAMES     6000
#define WT_STRIDE  66              // 64 + 2 pad: breaks LDS bank aliasing, keeps 8B align (even)
#define CHUNK      256
#define NCHUNKS    (T_SAMPLES / CHUNK)   // 3750
#define NTILES     (T_SAMPLES / 16)      // 60000 (tiles never straddle a 160-sample frame)

typedef float v2f __attribute__((ext_vector_type(2)));
typedef float v8f __attribute__((ext_vector_type(8)));

// ---- workspace layout (floats) ----
#define OFF_ATT    0                         // 6000*64 softmaxed attention, [frame][k]
#define OFF_WT     (FRAMES * N_TAB)          // 512*66 transposed/tanh'd tables, [pos][k]
#define OFF_CSUM   (OFF_WT + WT_LEN * WT_STRIDE)
#define OFF_PHASE  (OFF_CSUM + NCHUNKS)      // 960000 phase values
// total ~1.38M floats (~5.6 MB) of d_ws

// Build effective wavetable: tanh on tables 4..63, transposed to [pos][k] with pad stride.
__global__ void k_buildwt(const float* __restrict__ wt, float* __restrict__ wtsT) {
    int i = blockIdx.x * blockDim.x + threadIdx.x;
    if (i >= N_TAB * WT_LEN) return;
    int k = i >> 9, pos = i & (WT_LEN - 1);
    float v = wt[i];
    if (k >= 4) v = tanhf(v);
    wtsT[pos * WT_STRIDE + k] = v;
}

// Softmax over the 64 tables per frame. attention is [k][frame].
__global__ void k_softmax(const float* __restrict__ att, float* __restrict__ out) {
    int f = blockIdx.x * blockDim.x + threadIdx.x;
    if (f >= FRAMES) return;
    float mx = -3.4e38f;
    for (int k = 0; k < N_TAB; ++k) mx = fmaxf(mx, att[k * FRAMES + f]);
    float s = 0.f;
    for (int k = 0; k < N_TAB; ++k) s += expf(att[k * FRAMES + f] - mx);
    float inv = 1.0f / s;
    for (int k = 0; k < N_TAB; ++k) out[f * N_TAB + k] = expf(att[k * FRAMES + f] - mx) * inv;
}

// Blocked scan stage 1: per-256-sample chunk sums (sequential-order adds to mimic f32 cumsum).
__global__ void k_chunksum(const float* __restrict__ pitch, float* __restrict__ csum) {
    __shared__ float sh[CHUNK];
    int t = blockIdx.x * CHUNK + threadIdx.x;
    sh[threadIdx.x] = pitch[t] * (512.0f / 16000.0f);
    __syncthreads();
    if (threadIdx.x == 0) {
        float s = 0.f;
        for (int i = 0; i < CHUNK; ++i) s += sh[i];
        csum[blockIdx.x] = s;
    }
}

// Stage 2: serial exclusive scan of 3750 chunk sums.
__global__ void k_scanchunks(float* csum) {
    float run = 0.f;
    for (int i = 0; i < NCHUNKS; ++i) { float v = csum[i]; csum[i] = run; run += v; }
}

// Stage 3: exclusive scan within chunk (sequential rounding order) -> phase = cumsum mod 512.
__global__ void k_phase(const float* __restrict__ pitch, const float* __restrict__ csum,
                        float* __restrict__ phase) {
    __shared__ float sh[CHUNK];
    int t = blockIdx.x * CHUNK + threadIdx.x;
    sh[threadIdx.x] = pitch[t] * (512.0f / 16000.0f);
    __syncthreads();
    if (threadIdx.x == 0) {
        float run = csum[blockIdx.x];
        for (int i = 0; i < CHUNK; ++i) { float v = sh[i]; sh[i] = run; run += v; }
    }
    __syncthreads();
    phase[t] = fmodf(sh[threadIdx.x], 512.0f);
}

// Fused oscillator + attention mix + amplitude, reduction over 64 tables done with
// chained V_WMMA_F32_16X16X4_F32 (16 K-steps of 4). One wave produces 16 samples/tile.
__global__ void __launch_bounds__(256) k_synth(const float* __restrict__ amp,
                                               const float* __restrict__ wtsT,
                                               const float* __restrict__ attsm,
                                               const float* __restrict__ phase,
                                               float* __restrict__ out) {
    extern __shared__ float smem[];   // [WT_LEN*WT_STRIDE] table + 8*16 per-wave scratch

    // ---- stage the 135KB table into LDS with the async DMA path (ASYNCcnt) ----
    const int totalChunks = (WT_LEN * WT_STRIDE * 4) / 16;   // 8448 x b128
    unsigned ldsBase = (unsigned)(unsigned long long)(uintptr_t)smem;  // low 32b of flat = LDS offset
    for (int ch = threadIdx.x; ch < totalChunks; ch += blockDim.x) {
        unsigned la = ldsBase + (unsigned)ch * 16u;
        unsigned long long ga = (unsigned long long)(uintptr_t)wtsT + (unsigned long long)ch * 16ull;
        asm volatile("global_load_async_to_lds_b128 %0, %1, off" :: "v"(la), "v"(ga) : "memory");
    }
    asm volatile("s_wait_asynccnt 0" ::: "memory");
    __syncthreads();

    const float* ldsWT = smem;
    float* scratch = smem + WT_LEN * WT_STRIDE;

    const int lane = threadIdx.x & 31;
    const int wib  = threadIdx.x >> 5;
    const int gw   = blockIdx.x * (blockDim.x >> 5) + wib;
    const int nw   = gridDim.x * (blockDim.x >> 5);
    const int m    = lane & 15;            // sample row within tile
    const int koff = (lane >> 4) << 1;     // K sub-offset per WMMA register layout
    float* wsc = scratch + wib * 16;

    for (int tile = gw; tile < NTILES; tile += nw) {
        const int base = tile << 4;
        const int f = tile / 10;           // 160-sample frame index (tiles never straddle)
        const int t = base + m;

        const float idx = phase[t];
        const float lof = floorf(idx);
        const float alpha = idx - lof;
        int lo = (int)lof;
        int hi = (int)ceilf(idx);
        hi = (hi >= WT_LEN) ? 0 : hi;
        const int lobase = lo * WT_STRIDE + koff;
        const int hibase = hi * WT_STRIDE + koff;
        const float2* attp = (const float2*)(attsm + f * N_TAB);

        v8f c = {0.f, 0.f, 0.f, 0.f, 0.f, 0.f, 0.f, 0.f};
#pragma unroll
        for (int s = 0; s < 16; ++s) {
            // A[m, 4s+koff .. +1]: lerp two table values (ds_load_b64, padded banks)
            float2 wl = *(const float2*)(ldsWT + lobase + 4 * s);
            float2 wh = *(const float2*)(ldsWT + hibase + 4 * s);
            v2f a, b;
            a.x = wl.x + alpha * (wh.x - wl.x);
            a.y = wl.y + alpha * (wh.y - wl.y);
            // B[4s+koff .. +1, n] replicated across the 16 columns
            float2 bv = attp[2 * s + (koff >> 1)];
            b.x = bv.x; b.y = bv.y;
            c = __builtin_amdgcn_wmma_f32_16x16x4_f32(false, a, false, b, (short)0, c,
                                                      false, false);
        }

        // Every column of C is identical: rows 0..7 live in lane 0, rows 8..15 in lane 16.
        if (m == 0) {
            int roff = (lane >> 4) << 3;
#pragma unroll
            for (int r = 0; r < 8; ++r) wsc[roff + r] = c[r];
        }
        if (lane < 16) {
            out[t] = wsc[lane] * amp[t];   // fold in amplitude
        }
    }
}

extern "C" void kernel_launch(void* const* d_in, const int* in_sizes, int n_in,
                              void* d_out, int out_size, void* d_ws, size_t ws_size,
                              hipStream_t stream) {
    const float* pitch = (const float*)d_in[0];
    const float* amp   = (const float*)d_in[1];
    const float* wav   = (const float*)d_in[2];
    const float* att   = (const float*)d_in[3];
    float* out = (float*)d_out;
    float* ws  = (float*)d_ws;

    float* att_sm = ws + OFF_ATT;
    float* wtsT   = ws + OFF_WT;
    float* csum   = ws + OFF_CSUM;
    float* phase  = ws + OFF_PHASE;

    k_buildwt<<<(N_TAB * WT_LEN + 255) / 256, 256, 0, stream>>>(wav, wtsT);
    k_softmax<<<(FRAMES + 255) / 256, 256, 0, stream>>>(att, att_sm);
    k_chunksum<<<NCHUNKS, CHUNK, 0, stream>>>(pitch, csum);
    k_scanchunks<<<1, 1, 0, stream>>>(csum);
    k_phase<<<NCHUNKS, CHUNK, 0, stream>>>(pitch, csum, phase);

    const size_t smemBytes = (size_t)(WT_LEN * WT_STRIDE + 8 * 16) * sizeof(float); // ~135.7 KB
    k_synth<<<512, 256, smemBytes, stream>>>(amp, wtsT, att_sm, phase, out);
}